// EnergyGateMemory_35270271435232
// MI455X (gfx1250) — compile-verified
//
#include <hip/hip_runtime.h>
#include <math.h>

// ---------------- problem constants (match reference) ----------------
#define Bb 32
#define Tt 1024
#define Hh 512
#define Mm 128
#define Vv 64
#define TS (Tt - 3)                    // 1021 scan steps
#define TC 16                          // tokens per chunk
#define NCH ((TS + TC - 1) / TC)       // 64 chunks
#define MS (Hh + 4)                    // padded LDS row stride for memory
#define TSR Hh                         // dense LDS row stride for tokens (TDM tile)
#define EPSF 1e-8f

typedef float v2f __attribute__((ext_vector_type(2)));
typedef float v8f __attribute__((ext_vector_type(8)));
typedef unsigned int u32x4 __attribute__((ext_vector_type(4)));
typedef int i32x8 __attribute__((ext_vector_type(8)));
typedef int i32x4 __attribute__((ext_vector_type(4)));

// ---------------- kscan LDS layout (float offsets) ----------------
#define MEM_OFF 0                      // 128 x 516 memory state
#define TOK_OFF (Mm * MS)              // 16 x 512 chunk tokens (dense, TDM dest)
#define D_OFF   (TOK_OFF + TC * TSR)   // 128 x 16  D = mem . toks^T
#define G_OFF   (D_OFF + Mm * TC)      // 16 x 16   G = toks . toks^T
#define MSQ_OFF (G_OFF + TC * TC)      // 128       ||mem[m]||^2
#define LW_OFF  (MSQ_OFF + Mm)         // 128 ints  last_writer per slot
#define SMEM_FLOATS (LW_OFF + Mm)
#define SMEM_BYTES (SMEM_FLOATS * 4)   // 307,200 B <= 320 KB/WGP (CDNA5)

// =====================================================================
// Kernel 0: zero the global write counter
// =====================================================================
__global__ void kinit(int* cnt) { *cnt = 0; }

// =====================================================================
// Kernel 1: gate_probs = sigmoid(enc @ Wg + bg)   (B*T rows, 1 wave/row)
// =====================================================================
__global__ void kgate(const float* __restrict__ enc, const float* __restrict__ Wg,
                      const float* __restrict__ bg, float* __restrict__ gate_out) {
  int row  = blockIdx.x * 8 + (threadIdx.x >> 5);
  int lane = threadIdx.x & 31;
  if (row >= Bb * Tt) return;
  const float4* e = (const float4*)(enc + (size_t)row * Hh);
  const float4* w = (const float4*)Wg;
  float s = 0.f;
  for (int j = lane; j < Hh / 4; j += 32) {
    float4 a = e[j], b = w[j];
    s += a.x * b.x + a.y * b.y + a.z * b.z + a.w * b.w;
  }
  for (int off = 16; off > 0; off >>= 1) s += __shfl_down(s, off, 32);
  if (lane == 0) gate_out[row] = 1.f / (1.f + expf(-(s + bg[0])));
}

// =====================================================================
// Kernel 2: sequential energy-gated memory scan.
// One workgroup per batch; 300KB of memory state resident in CDNA5 LDS.
// Chunk tokens are staged by the Tensor Data Mover (tensor_load_to_lds,
// TENSORcnt-tracked); per chunk, D = mem(128x512)@toks^T and the Gram
// G = toks@toks^T are computed with V_WMMA_F32_16X16X4_F32; one wave
// then replays the 16 steps using D/G + last_writer indirection.
// =====================================================================
__global__ void __launch_bounds__(256)
kscan(const float* __restrict__ enc, const float* __restrict__ gates,
      float* __restrict__ mem_out, int* __restrict__ cnt) {
  extern __shared__ float smem[];
  float* memL = smem + MEM_OFF;
  float* tokL = smem + TOK_OFF;
  float* Dm   = smem + D_OFF;
  float* Gm   = smem + G_OFF;
  float* msq  = smem + MSQ_OFF;
  int*   lw   = (int*)(smem + LW_OFF);

  const int b    = blockIdx.x;
  const int tid  = threadIdx.x;
  const int wave = tid >> 5;
  const int lane = tid & 31;
  const int n    = lane & 15;       // WMMA: A row / B,C,D column index
  const int hi   = lane >> 4;       // WMMA: half-wave selector

  // ---- init: memory0 = zeros, ||mem||^2 = 0, last_writer = -1 ----
  for (int x = tid; x < Mm * MS; x += 256) memL[x] = 0.f;
  for (int x = tid; x < Mm; x += 256) { msq[x] = 0.f; lw[x] = -1; }
  __syncthreads();

  int nwrites = 0;  // valid in wave 0 lane 0

  for (int c = 0; c < NCH; ++c) {
    const float* src = enc + ((size_t)b * Tt + (size_t)c * TC) * Hh;

#if __has_builtin(__builtin_amdgcn_tensor_load_to_lds)
    // ---- TDM: DMA the 16x512 f32 token tile into LDS (TENSORcnt) ----
    if (wave == 0) {
      unsigned long long ga = (unsigned long long)(const void*)src;
      u32x4 g0;
      g0[0] = 1u;                                        // count=1 (valid D#)
      g0[1] = (unsigned)(TOK_OFF * 4);                   // lds byte address
      g0[2] = (unsigned)(ga & 0xFFFFFFFFull);            // global_addr[31:0]
      g0[3] = (unsigned)((ga >> 32) & 0x1FFFFFFull)      // global_addr[56:32]
            | 0x80000000u;                               // type=2 (image)
      i32x8 g1;
      g1[0] = 0x00020000;                                // data_size=4B
      g1[1] = (int)((unsigned)(Hh & 0xFFFF) << 16);      // tensor_dim0 lo16 = 512
      g1[2] = (int)((unsigned)(TC & 0xFFFF) << 16);      // tensor_dim1 lo16 = 16
      g1[3] = (int)((unsigned)(Hh & 0xFFFF) << 16);      // tile_dim0 = 512
      g1[4] = TC;                                        // tile_dim1 = 16
      g1[5] = Hh;                                        // tensor_dim0_stride = 512
      g1[6] = 0; g1[7] = 0;
      i32x4 g2 = {0, 0, 0, 0}, g3 = {0, 0, 0, 0};        // <=2D tensor
      i32x8 g4 = {0, 0, 0, 0, 0, 0, 0, 0};               // extended group (unused)
      __builtin_amdgcn_tensor_load_to_lds(g0, g1, g2, g3, g4, 0);
      __builtin_amdgcn_s_wait_tensorcnt(0);
    }
#else
    // ---- fallback: cooperative float4 staging ----
    for (int x = tid; x < TC * (Hh / 4); x += 256) {
      int i = x >> 7, h4 = x & 127;
      *(float4*)(tokL + i * TSR + h4 * 4) = ((const float4*)src)[x];
    }
#endif
    // prefetch next chunk's rows into GL2 while we compute (gfx1250 path)
    if (c + 1 < NCH && lane == 0)
      __builtin_prefetch(enc + ((size_t)b * Tt + (size_t)(c + 1) * TC + wave * 2) * Hh, 0, 0);
    __syncthreads();

    // ---- WMMA GEMMs: jobs 0..7 = D tiles (16 rows each), job 8 = G ----
    for (int job = wave; job < 9; job += 8) {
      const float* Arow = (job < 8) ? (memL + (size_t)(job * 16 + n) * MS)
                                    : (tokL + (size_t)n * TSR);
      const float* Brow = tokL + (size_t)n * TSR;        // B[k][n] = toks[n][k]
      v8f acc = {0.f, 0.f, 0.f, 0.f, 0.f, 0.f, 0.f, 0.f};
      for (int k0 = 0; k0 < Hh; k0 += 4) {
        int kk = k0 + (hi << 1);                         // upper half-wave: K+2
        v2f av, bv;
        av.x = Arow[kk]; av.y = Arow[kk + 1];
        bv.x = Brow[kk]; bv.y = Brow[kk + 1];
        acc = __builtin_amdgcn_wmma_f32_16x16x4_f32(
            false, av, false, bv, (short)0, acc, false, false);
      }
      // C/D layout: VGPR r holds row (r + 8*hi), column = lane&15
      if (job < 8) {
        for (int r = 0; r < 8; ++r)
          Dm[(job * 16 + r + 8 * hi) * TC + n] = acc[r];
      } else {
        for (int r = 0; r < 8; ++r)
          Gm[(r + 8 * hi) * TC + n] = acc[r];
      }
    }
    __syncthreads();

    // ---- serial walk of the chunk (wave 0 only; 4 slots per lane) ----
    if (wave == 0) {
      for (int i = 0; i < TC; ++i) {
        int s = c * TC + i;
        if (s >= TS) break;
        float tsq = Gm[i * TC + i];                      // ||tok||^2
        float bestv = 3.0e38f; int besti = 0;
        for (int j = 0; j < 4; ++j) {
          int m = lane * 4 + j;                          // ascending => first-min
          int w = lw[m];
          float d     = (w < 0) ? Dm[m * TC + i] : Gm[w * TC + i];
          float denom = fmaxf(sqrtf(msq[m]) * sqrtf(tsq), EPSF);
          float sim   = d / denom;
          if (sim < bestv) { bestv = sim; besti = m; }
        }
        for (int off = 16; off > 0; off >>= 1) {         // argmin, ties -> low idx
          float ov = __shfl_down(bestv, off, 32);
          int   oi = __shfl_down(besti, off, 32);
          if (ov < bestv || (ov == bestv && oi < besti)) { bestv = ov; besti = oi; }
        }
        besti = __shfl(besti, 0, 32);
        float bsq  = msq[besti];
        float gate = gates[(size_t)b * Tt + s];
        bool  dow  = (tsq > bsq) && (gate > 0.5f);
        if (dow && lane == 0) {
          lw[besti]  = i;
          msq[besti] = tsq;
          ++nwrites;
        }
      }
    }
    __syncthreads();

    // ---- apply chunk writes to the LDS memory state, reset lw ----
    for (int m = 0; m < Mm; ++m) {
      int w = lw[m];
      if (w >= 0) {
        for (int x = tid; x < Hh / 4; x += 256)
          *(float4*)(memL + (size_t)m * MS + x * 4) =
              *(float4*)(tokL + (size_t)w * TSR + x * 4);
      }
    }
    __syncthreads();
    if (tid < Mm) lw[tid] = -1;
    __syncthreads();
  }

  // ---- spill final memory state for the attention kernel ----
  float* dst = mem_out + (size_t)b * Mm * Hh;
  for (int x = tid; x < Mm * (Hh / 4); x += 256) {
    int m = x >> 7, h4 = x & 127;
    ((float4*)dst)[x] = *(float4*)(memL + (size_t)m * MS + h4 * 4);
  }
  if (tid == 0) atomicAdd(cnt, nwrites);
}

// =====================================================================
// Kernel 3: attention readout. scores = mem @ (Wk^T q) + q.bk  (avoids
// materializing k = mem@Wk entirely), softmax, retrieve, project.
// =====================================================================
__global__ void kattn(const float* __restrict__ query,
                      const float* __restrict__ Wq, const float* __restrict__ bq,
                      const float* __restrict__ Wk, const float* __restrict__ bk,
                      const float* __restrict__ Wo, const float* __restrict__ bo,
                      const float* __restrict__ memg, float* __restrict__ logits) {
  __shared__ float qrow[Hh], qh[Hh], vh[Hh], sc[Mm], red[1];
  const int b = blockIdx.x, tid = threadIdx.x;

  for (int h = tid; h < Hh; h += 256) qrow[h] = query[(size_t)b * Hh + h];
  __syncthreads();

  for (int h = tid; h < Hh; h += 256) {              // q = query@Wq + bq
    float s = bq[h];
    for (int d = 0; d < Hh; ++d) s += qrow[d] * Wq[(size_t)d * Hh + h];
    qh[h] = s;
  }
  __syncthreads();

  for (int h = tid; h < Hh; h += 256) {              // v = Wk^T q  (row-major dot)
    float s = 0.f;
    const float* wr = Wk + (size_t)h * Hh;
    for (int d = 0; d < Hh; ++d) s += wr[d] * qh[d];
    vh[h] = s;
  }
  float qb = 0.f;                                    // q . bk (redundant/thread)
  for (int d = 0; d < Hh; ++d) qb += qh[d] * bk[d];
  __syncthreads();

  const float* mb = memg + (size_t)b * Mm * Hh;
  if (tid < Mm) {
    const float* mr = mb + (size_t)tid * Hh;
    float s = 0.f;
    for (int d = 0; d < Hh; ++d) s += mr[d] * vh[d];
    sc[tid] = (s + qb) * 0.04419417382415922f;       // 1/sqrt(512)
  }
  __syncthreads();
  if (tid == 0) {
    float mx = sc[0];
    for (int m = 1; m < Mm; ++m) mx = fmaxf(mx, sc[m]);
    float sum = 0.f;
    for (int m = 0; m < Mm; ++m) { sc[m] = expf(sc[m] - mx); sum += sc[m]; }
    red[0] = 1.f / sum;
  }
  __syncthreads();
  float inv = red[0];
  for (int h = tid; h < Hh; h += 256) {              // retrieved + query
    float s = 0.f;
    for (int m = 0; m < Mm; ++m) s += sc[m] * mb[(size_t)m * Hh + h];
    qh[h] = s * inv + qrow[h];
  }
  __syncthreads();
  if (tid < Vv) {                                    // logits
    float s = bo[tid];
    for (int h = 0; h < Hh; ++h) s += qh[h] * Wo[(size_t)h * Vv + tid];
    logits[b * Vv + tid] = s;
  }
}

// =====================================================================
// Kernel 4: write_rate
// =====================================================================
__global__ void kfinal(const int* cnt, float* wrate) {
  *wrate = (float)(*cnt) / (float)(Bb * TS);
}

// =====================================================================
extern "C" void kernel_launch(void* const* d_in, const int* in_sizes, int n_in,
                              void* d_out, int out_size, void* d_ws, size_t ws_size,
                              hipStream_t stream) {
  const float* enc   = (const float*)d_in[0];
  const float* query = (const float*)d_in[1];
  const float* Wg    = (const float*)d_in[2];
  const float* bg    = (const float*)d_in[3];
  const float* Wq    = (const float*)d_in[4];
  const float* bq    = (const float*)d_in[5];
  const float* Wk    = (const float*)d_in[6];
  const float* bk    = (const float*)d_in[7];
  const float* Wo    = (const float*)d_in[8];
  const float* bo    = (const float*)d_in[9];

  float* out    = (float*)d_out;
  float* logits = out;                       // (B,V)   = 2048
  float* gates  = out + Bb * Vv;             // (B,T)   = 32768
  float* wrate  = out + Bb * Vv + Bb * Tt;   // scalar

  int*   cnt   = (int*)d_ws;
  float* memws = (float*)((char*)d_ws + 256);  // (B,M,H) final memory state

  // allow > 64KB dynamic LDS for the scan kernel (CDNA5: 320KB/WGP)
  (void)hipFuncSetAttribute((const void*)kscan,
                            hipFuncAttributeMaxDynamicSharedMemorySize,
                            SMEM_BYTES);

  kinit<<<1, 1, 0, stream>>>(cnt);
  kgate<<<(Bb * Tt) / 8, 256, 0, stream>>>(enc, Wg, bg, gates);
  kscan<<<Bb, 256, SMEM_BYTES, stream>>>(enc, gates, memws, cnt);
  kattn<<<Bb, 256, 0, stream>>>(query, Wq, bq, Wk, bk, Wo, bo, memws, logits);
  kfinal<<<1, 1, 0, stream>>>(cnt, wrate);
}